// MoE_84765474554514
// MI455X (gfx1250) — compile-verified
//
#include <hip/hip_runtime.h>

// ---------------------------------------------------------------------------
// MoE (top-1) forward for MI455X / gfx1250.
//   - fp32 matmuls on V_WMMA_F32_16X16X4_F32 (faithful to fp32 reference)
//   - token gather x-tile staged by the Tensor Data Mover (gather mode,
//     16-bit row indices, LDS padding by TDM for bank-conflict-free reads)
//   - M-tile = 32 tokens: every weight (B) fetch feeds 2 WMMAs
//   inputs : x[4,2048,768], W_router[4,768], W_fc[4,3072,768], W_proj[4,768,3072]
//   outputs: y[4,2048,768] ++ aux ++ z_loss ++ router_entropy ++ actual[4]
// ---------------------------------------------------------------------------

#define USE_TDM 1

namespace {
constexpr int kBT = 4 * 2048;       // 8192 tokens
constexpr int kC  = 768;
constexpr int kH  = 3072;
constexpr int kE  = 4;
constexpr int kYN = kBT * kC;       // 6291456
constexpr int kRouterBlocks = kBT / 8;     // 8 waves (tokens) per block
// workspace layout (dwords):
constexpr int kPermOff = 16;                         // int perm[E][BT]
constexpr int kPartOff = kPermOff + kE * kBT;        // float partials[blocks][6]

constexpr int kM = 32;              // tokens per workgroup (2 x 16-row tiles)
constexpr int kHChunk = 128;        // h-dimension chunk held in LDS
// TDM pads 4 dwords after every 256 dwords stored -> x row stride 768+12=780.
// 780 mod 64 == 12 -> 16 A-rows hit 16 distinct banks; kb halves disjoint.
constexpr int kXStride = kC + 12;   // 780 dwords
constexpr int kHPad = kHChunk + 4;  // 132 (mod 64 == 4 -> conflict-free)
constexpr int kXBytes = kM * kXStride * 4;   // 99840
constexpr int kHBytes = kM * kHPad * 4;      // 16896
constexpr int kLdsBytes = kXBytes + kHBytes + kM * 4;   // ~116.8 KB dynamic LDS
}

typedef __attribute__((ext_vector_type(2))) float v2f;
typedef __attribute__((ext_vector_type(8))) float v8f;
typedef __attribute__((ext_vector_type(4))) unsigned int u32x4;
typedef __attribute__((ext_vector_type(8))) int i32x8;
typedef __attribute__((ext_vector_type(4))) int i32x4;

// D(16x16,f32) = A(16x4,f32) * B(4x16,f32) + C
// A: lane -> m = lane&15 ; VGPR0/1 = K = (lane>>4)*2 + {0,1}
// B: lane -> n = lane&15 ; VGPR0/1 = K = (lane>>4)*2 + {0,1}
// C/D: VGPR r, lanes 0-15 -> M=r, N=lane ; lanes 16-31 -> M=r+8, N=lane-16
__device__ __forceinline__ v8f wmma_f32_16x16x4(v2f a, v2f b, v8f c) {
  return __builtin_amdgcn_wmma_f32_16x16x4_f32(false, a, false, b, (short)0, c,
                                               false, false);
}

#if USE_TDM
// TDM gather-mode load: up to 16 rows of x (768 f32 each), rows selected by
// 16-bit token indices, padded into LDS (4 dwords per 256 stored).
__device__ __forceinline__ void tdm_gather_rows16(const float* tensor_base,
                                                  unsigned lds_byte_addr,
                                                  const int* rows, int start,
                                                  int cnt_e) {
  const unsigned nval = (unsigned)min(16, cnt_e - start);
  const unsigned long long ga = (unsigned long long)(uintptr_t)tensor_base;
  u32x4 g0;
  g0[0] = 0x80000001u;  // gather_mode=1, 16-bit indices, count=1 descriptor
  g0[1] = lds_byte_addr;
  g0[2] = (unsigned)(ga & 0xFFFFFFFFu);
  g0[3] = (unsigned)((ga >> 32) & 0x01FFFFFFu) | (2u << 30);  // type=2 "image"
  i32x8 g1;
  // data_size=4B(2), pad_enable, pad_interval=256dw(7), pad_amount=4dw(3)
  g1[0] = (int)0x07D20000u;
  g1[1] = (int)(((unsigned)kC & 0xFFFFu) << 16);                 // dim0 lo16
  g1[2] = (int)(((unsigned)kBT & 0xFFFFu) << 16);                // dim1 lo16
  g1[3] = (int)(((unsigned)kC & 0xFFFFu) << 16);                 // tile_dim0
  g1[4] = (int)nval;                                             // tile_dim1
  g1[5] = (int)kC;                                               // dim0 stride
  g1[6] = 0;
  g1[7] = 0;
  unsigned w[8];
#pragma unroll
  for (int i = 0; i < 8; ++i) {  // pack 16-bit row indices (token ids)
    const int s0 = start + 2 * i, s1 = s0 + 1;
    const unsigned t0 = (s0 < cnt_e) ? (unsigned)rows[s0] : 0u;
    const unsigned t1 = (s1 < cnt_e) ? (unsigned)rows[s1] : 0u;
    w[i] = (t0 & 0xFFFFu) | ((t1 & 0xFFFFu) << 16);
  }
  i32x4 g2 = {(int)w[0], (int)w[1], (int)w[2], (int)w[3]};
  i32x4 g3 = {(int)w[4], (int)w[5], (int)w[6], (int)w[7]};
#if defined(__clang_major__) && (__clang_major__ >= 23)
  i32x8 g4 = {0, 0, 0, 0, 0, 0, 0, 0};
  __builtin_amdgcn_tensor_load_to_lds(g0, g1, g2, g3, g4, 0);
#else
  __builtin_amdgcn_tensor_load_to_lds(g0, g1, g2, g3, 0);
#endif
}
#endif  // USE_TDM

// ---------------------------------------------------------------------------
__global__ void moe_init_kernel(int* __restrict__ counts) {
  if (threadIdx.x < 16) counts[threadIdx.x] = 0;
}

// ---------------------------------------------------------------------------
// Router: wave-per-token; logits, softmax stats, argmax dispatch into perm.
// ---------------------------------------------------------------------------
__global__ __launch_bounds__(256) void moe_router_kernel(
    const float* __restrict__ x, const float* __restrict__ Wr,
    int* __restrict__ counts, int* __restrict__ perm,
    float* __restrict__ partials) {
  __shared__ float red[8][8];
  const int tid  = threadIdx.x;
  const int wave = tid >> 5;
  const int lane = tid & 31;
  const int t    = blockIdx.x * 8 + wave;

  float p0 = 0.f, p1 = 0.f, p2 = 0.f, p3 = 0.f;
  if (t < kBT) {
    const float* xr = x + (size_t)t * kC;
    for (int c = lane; c < kC; c += 32) {
      const float xv = xr[c];
      p0 += xv * Wr[0 * kC + c];
      p1 += xv * Wr[1 * kC + c];
      p2 += xv * Wr[2 * kC + c];
      p3 += xv * Wr[3 * kC + c];
    }
  }
#pragma unroll
  for (int off = 16; off >= 1; off >>= 1) {
    p0 += __shfl_xor(p0, off, 32);
    p1 += __shfl_xor(p1, off, 32);
    p2 += __shfl_xor(p2, off, 32);
    p3 += __shfl_xor(p3, off, 32);
  }

  if (lane == 0) {
    float l[4] = {p0, p1, p2, p3};
    float m = l[0];
    int ei = 0;
#pragma unroll
    for (int e = 1; e < 4; ++e)
      if (l[e] > m) { m = l[e]; ei = e; }   // strict > : first-index tie-break
    float pr[4], s = 0.f;
#pragma unroll
    for (int e = 0; e < 4; ++e) { pr[e] = __expf(l[e] - m); s += pr[e]; }
    const float inv = 1.f / s;
    float hent = 0.f;
#pragma unroll
    for (int e = 0; e < 4; ++e) {
      pr[e] *= inv;
      hent -= pr[e] * __logf(pr[e] + 1e-9f);
    }
    const float lse = m + __logf(s);
    if (t < kBT) {
      const int pos = atomicAdd(&counts[ei], 1);  // slot in expert list
      perm[ei * kBT + pos] = t;
    }
    red[wave][0] = pr[0]; red[wave][1] = pr[1];
    red[wave][2] = pr[2]; red[wave][3] = pr[3];
    red[wave][4] = lse * lse;
    red[wave][5] = hent;
  }
  __syncthreads();
  if (tid < 6) {
    float s = 0.f;
#pragma unroll
    for (int w = 0; w < 8; ++w) s += red[w][tid];
    partials[blockIdx.x * 6 + tid] = s;
  }
}

// ---------------------------------------------------------------------------
__global__ void moe_finalize_kernel(const int* __restrict__ counts,
                                    const float* __restrict__ partials,
                                    float* __restrict__ out_scal) {
  const int lane = threadIdx.x;
  float a[6] = {0.f, 0.f, 0.f, 0.f, 0.f, 0.f};
  for (int b = lane; b < kRouterBlocks; b += 32)
#pragma unroll
    for (int j = 0; j < 6; ++j) a[j] += partials[b * 6 + j];
#pragma unroll
  for (int off = 16; off >= 1; off >>= 1)
#pragma unroll
    for (int j = 0; j < 6; ++j) a[j] += __shfl_xor(a[j], off, 32);

  if (lane == 0) {
    const float invBT = 1.f / (float)kBT;
    float aux = 0.f, act[4];
#pragma unroll
    for (int e = 0; e < 4; ++e) {
      act[e] = (float)counts[e] * invBT;
      aux += act[e] * (a[e] * invBT);
    }
    out_scal[0] = (float)kE * aux;
    out_scal[1] = a[4] * invBT;
    out_scal[2] = (a[5] * invBT) * (1.f / 1.3862943611198906f);
#pragma unroll
    for (int e = 0; e < 4; ++e) out_scal[3 + e] = act[e];
  }
}

// ---------------------------------------------------------------------------
// Expert kernel: one workgroup = one expert x 32 gathered tokens. 8 waves.
// Fused y = Wproj * relu(Wfc * x)^2, H chunked 128-wide through LDS.
// Each weight float2 load feeds two WMMAs (row-halves of the M-tile).
// ---------------------------------------------------------------------------
__global__ __launch_bounds__(256) void moe_expert_gemm_kernel(
    const float* __restrict__ x,
    const float* __restrict__ Wfc,    // [E,H,C]
    const float* __restrict__ Wproj,  // [E,C,H]
    const int* __restrict__ counts,
    const int* __restrict__ perm,
    float* __restrict__ y) {
  extern __shared__ char smem[];
  float* lds_x = (float*)smem;                       // 32 x 780 dwords
  float* lds_h = (float*)(smem + kXBytes);           // 32 x 132 dwords
  int*   tokv  = (int*)(smem + kXBytes + kHBytes);   // 32 tokens

  const int e    = blockIdx.y;
  const int tile = blockIdx.x;
  const int cnt  = counts[e];
  const int base = tile * kM;
  if (base >= cnt) return;              // uniform early-exit

  const int tid  = threadIdx.x;
  const int wave = tid >> 5;
  const int lane = tid & 31;
  const int m16  = lane & 15;
  const int kb   = (lane >> 4) << 1;    // K sub-offset 0 or 2
  const int mhi  = (lane >> 4) << 3;    // C/D row offset 0 or 8

  const int* rows = perm + e * kBT;
  if (tid < kM) {
    const int s = base + tid;
    tokv[tid] = (s < cnt) ? rows[s] : -1;   // -1 = pad slot (never stored)
  }

#if USE_TDM
  if (wave == 0) {   // TDM ignores EXEC; issue once per workgroup
    tdm_gather_rows16(x, (unsigned)(uintptr_t)lds_x, rows, base, cnt);
    if (cnt - base > 16)
      tdm_gather_rows16(x, (unsigned)(uintptr_t)(lds_x + 16 * kXStride), rows,
                        base + 16, cnt);
    __builtin_amdgcn_s_wait_tensorcnt(0);
  }
#else
  for (int i = tid; i < kM * (kC / 4); i += 256) {
    const int row = i / (kC / 4);
    const int col = (i % (kC / 4)) * 4;
    const int s = base + row;
    float4 v = make_float4(0.f, 0.f, 0.f, 0.f);
    if (s < cnt) v = *(const float4*)(x + (size_t)rows[s] * kC + col);
    *(float4*)(&lds_x[row * kXStride + col + ((col >> 8) << 2)]) = v;
  }
#endif
  __syncthreads();

  const float* wfc = Wfc   + (size_t)e * kH * kC;
  const float* wpj = Wproj + (size_t)e * kC * kH;

  v8f accA[6], accB[6];
#pragma unroll
  for (int ti = 0; ti < 6; ++ti) { accA[ti] = (v8f){}; accB[ti] = (v8f){}; }
  const int n0p = wave * 96;

  for (int hc = 0; hc < kH; hc += kHChunk) {
    __syncthreads();  // lds_h fully consumed by previous GEMM2

    // --- GEMM1: act[32][wave*16 .. +16), K = 768 (3 x 256-dword segments)
    {
      const float* wb = wfc + (size_t)(hc + wave * 16 + m16) * kC + kb;
      v8f h0 = (v8f){}, h1 = (v8f){};
#pragma unroll
      for (int cb = 0; cb < kC; cb += 256) {
        const float* la  = lds_x + m16 * kXStride + cb + (cb >> 6) + kb;
        const float* lb  = la + 16 * kXStride;
        const float* wbc = wb + cb;
        for (int c0 = 0; c0 < 256; c0 += 4) {
          const v2f b  = *(const v2f*)(wbc + c0);
          const v2f a0 = *(const v2f*)(la + c0);
          const v2f a1 = *(const v2f*)(lb + c0);
          h0 = wmma_f32_16x16x4(a0, b, h0);
          h1 = wmma_f32_16x16x4(a1, b, h1);
        }
      }
#pragma unroll
      for (int r = 0; r < 8; ++r) {   // relu^2, scatter into act chunk
        float v0 = h0[r]; v0 = (v0 > 0.f) ? v0 : 0.f;
        float v1 = h1[r]; v1 = (v1 > 0.f) ? v1 : 0.f;
        lds_h[(r + mhi) * kHPad + wave * 16 + m16]      = v0 * v0;
        lds_h[(16 + r + mhi) * kHPad + wave * 16 + m16] = v1 * v1;
      }
    }
    __syncthreads();  // act chunk visible to all waves

    // --- GEMM2: accumulate y[32][wave*96 .. +96) over this h chunk
#pragma unroll
    for (int ti = 0; ti < 6; ++ti) {
      const float* wb = wpj + (size_t)(n0p + ti * 16 + m16) * kH + hc + kb;
      const float* la = lds_h + m16 * kHPad + kb;
      const float* lb = la + 16 * kHPad;
      for (int k0 = 0; k0 < kHChunk; k0 += 4) {
        const v2f b  = *(const v2f*)(wb + k0);
        const v2f a0 = *(const v2f*)(la + k0);
        const v2f a1 = *(const v2f*)(lb + k0);
        accA[ti] = wmma_f32_16x16x4(a0, b, accA[ti]);
        accB[ti] = wmma_f32_16x16x4(a1, b, accB[ti]);
      }
    }
  }

  // Scatter y rows to their tokens (pad slots skipped).
#pragma unroll
  for (int ti = 0; ti < 6; ++ti) {
    const int n0 = n0p + ti * 16;
#pragma unroll
    for (int r = 0; r < 8; ++r) {
      const int t0 = tokv[r + mhi];
      if (t0 >= 0) y[(size_t)t0 * kC + n0 + m16] = accA[ti][r];
      const int t1 = tokv[16 + r + mhi];
      if (t1 >= 0) y[(size_t)t1 * kC + n0 + m16] = accB[ti][r];
    }
  }
}

// ---------------------------------------------------------------------------
extern "C" void kernel_launch(void* const* d_in, const int* in_sizes, int n_in,
                              void* d_out, int out_size, void* d_ws,
                              size_t ws_size, hipStream_t stream) {
  (void)in_sizes; (void)n_in; (void)out_size; (void)ws_size;
  const float* x   = (const float*)d_in[0];
  const float* Wr  = (const float*)d_in[1];
  const float* Wfc = (const float*)d_in[2];
  const float* Wpj = (const float*)d_in[3];
  float* out = (float*)d_out;

  int*   counts   = (int*)d_ws;
  int*   perm     = (int*)d_ws + kPermOff;
  float* partials = (float*)d_ws + kPartOff;

  moe_init_kernel<<<1, 32, 0, stream>>>(counts);
  moe_router_kernel<<<kRouterBlocks, 256, 0, stream>>>(x, Wr, counts, perm,
                                                       partials);
  dim3 grid(kBT / kM, kE);
  moe_expert_gemm_kernel<<<grid, 256, kLdsBytes, stream>>>(x, Wfc, Wpj, counts,
                                                           perm, out);
  moe_finalize_kernel<<<1, 32, 0, stream>>>(counts, partials, out + kYN);
}